// ActivationGINLayer_77567109366534
// MI455X (gfx1250) — compile-verified
//
#include <hip/hip_runtime.h>
#include <hip/hip_bf16.h>

typedef float v2f __attribute__((ext_vector_type(2)));
typedef float v8f __attribute__((ext_vector_type(8)));

#define D_FEAT 128
#define BN_EPS 1e-5f

// -------- Kernel 1: hn = h * norm  (into d_out), zero neigh + stats --------
__global__ void k1_prescale_zero(const float* __restrict__ h,
                                 const float* __restrict__ norm,
                                 float* __restrict__ hn,
                                 float* __restrict__ neigh,
                                 float* __restrict__ stats,
                                 int n4 /* total elems / 4 */) {
    if (blockIdx.x == 0 && threadIdx.x < 2 * D_FEAT) stats[threadIdx.x] = 0.0f;
    const float4 z = make_float4(0.f, 0.f, 0.f, 0.f);
    int stride = gridDim.x * blockDim.x;
    for (int i = blockIdx.x * blockDim.x + threadIdx.x; i < n4; i += stride) {
        float4 hv = ((const float4*)h)[i];
        float nv = norm[i >> 5];          // 32 float4 per row of 128
        float4 r;
        r.x = hv.x * nv; r.y = hv.y * nv; r.z = hv.z * nv; r.w = hv.w * nv;
        ((float4*)hn)[i]    = r;
        ((float4*)neigh)[i] = z;
    }
}

// -------- Kernel 2: neigh[dst] += hn[src]  (wave-per-edge-pair) --------
// Edge indices are wave-uniform: readfirstlane forces them into SGPRs so the
// src/dst loads lower to scalar s_load_b32 (one SMEM request per wave instead
// of 32 lane requests). 2-edge unroll issues both float4 gathers before the
// 8 atomics for MLP; global_prefetch_b8 streams the next edge-list chunk.
__global__ void k2_edge_scatter(const float* __restrict__ hn,
                                const int* __restrict__ src,
                                const int* __restrict__ dst,
                                float* __restrict__ neigh,
                                int nedges) {
    const int lane = threadIdx.x & 31;
    int wid = (blockIdx.x * blockDim.x + threadIdx.x) >> 5;
    wid = __builtin_amdgcn_readfirstlane(wid);       // SGPR wave id
    const int nwaves = (gridDim.x * blockDim.x) >> 5;
    const int stride = nwaves * 2;

    for (int e = wid * 2; e < nedges; e += stride) {
        // speculative prefetch of next chunk of the edge stream (dropped if OOB)
        __builtin_prefetch(src + e + stride, 0, 1);
        __builtin_prefetch(dst + e + stride, 0, 1);

        const bool has2 = (e + 1 < nedges);          // wave-uniform branch
        int s0 = src[e];                             // scalar loads (uniform addr)
        int d0 = dst[e];
        float4 v0 = ((const float4*)(hn + (size_t)s0 * D_FEAT))[lane];

        float4 v1;
        int s1 = 0, d1 = 0;
        if (has2) {
            s1 = src[e + 1];
            d1 = dst[e + 1];
            v1 = ((const float4*)(hn + (size_t)s1 * D_FEAT))[lane];
        }

        float* b0 = neigh + (size_t)d0 * D_FEAT + lane * 4;
        atomicAdd(b0 + 0, v0.x);
        atomicAdd(b0 + 1, v0.y);
        atomicAdd(b0 + 2, v0.z);
        atomicAdd(b0 + 3, v0.w);

        if (has2) {
            float* b1 = neigh + (size_t)d1 * D_FEAT + lane * 4;
            atomicAdd(b1 + 0, v1.x);
            atomicAdd(b1 + 1, v1.y);
            atomicAdd(b1 + 2, v1.z);
            atomicAdd(b1 + 3, v1.w);
        }
    }
}

// -------- Kernel 3: out = ((1+eps)*hn + neigh)*norm, fused f32-WMMA column stats --------
// Block = 256 threads = 8 waves; wave w owns feature slice [w*16, w*16+15].
// Per loop step a wave consumes 4 rows x 16 cols as the B operand of
// V_WMMA_F32_16X16X4_F32 with A = ones(16x4): D[m][n] = sum_k B[k][n],
// i.e. exact f32 column sums on the tensor pipe. One WMMA for sum(x),
// one for sum(x^2). EXEC stays all-ones (uniform trip counts, N % 4 == 0).
__global__ void k3_combine_stats(const float* __restrict__ hn,
                                 const float* __restrict__ neigh,
                                 const float* __restrict__ norm,
                                 const float* __restrict__ eps_p,
                                 float* __restrict__ out,
                                 float* __restrict__ stats,
                                 int nrows) {
    const int lane    = threadIdx.x & 31;
    const int wave    = threadIdx.x >> 5;            // 0..7 -> column slice
    const int col     = wave * 16 + (lane & 15);
    const int kOff    = (lane >> 4) * 2;             // lanes 0-15: K=0,1 ; lanes 16-31: K=2,3
    const float epsw  = 1.0f + eps_p[0];

    v2f a; a.x = 1.0f; a.y = 1.0f;                   // all-ones A (16x4)
    v8f accS = {};                                   // sum(x)   accumulator
    v8f accQ = {};                                   // sum(x^2) accumulator

    const int step = gridDim.x * 4;
    for (int r0 = blockIdx.x * 4; r0 + 3 < nrows; r0 += step) {
        int rx = r0 + kOff;
        int ry = rx + 1;
        float nx = norm[rx], ny = norm[ry];
        float hx = hn[(size_t)rx * D_FEAT + col];
        float hy = hn[(size_t)ry * D_FEAT + col];
        float gx = neigh[(size_t)rx * D_FEAT + col];
        float gy = neigh[(size_t)ry * D_FEAT + col];
        float ox = (epsw * hx + gx) * nx;
        float oy = (epsw * hy + gy) * ny;
        out[(size_t)rx * D_FEAT + col] = ox;         // in-place overwrite of hn buffer
        out[(size_t)ry * D_FEAT + col] = oy;

        v2f b;  b.x  = ox;      b.y  = oy;
        v2f b2; b2.x = ox * ox; b2.y = oy * oy;
        accS = __builtin_amdgcn_wmma_f32_16x16x4_f32(false, a, false, b,
                                                     (short)0, accS, false, false);
        accQ = __builtin_amdgcn_wmma_f32_16x16x4_f32(false, a, false, b2,
                                                     (short)0, accQ, false, false);
    }
    // D row M=0 lives in VGPR0 of lanes 0-15 (N = lane); column sum is
    // replicated across all M, so accS[0]/accQ[0] of lanes 0-15 suffice.
    if (lane < 16) {
        atomicAdd(&stats[col],          accS[0]);
        atomicAdd(&stats[D_FEAT + col], accQ[0]);
    }
}

// -------- Kernel 4: BatchNorm finalize + ReLU, in place on d_out --------
__global__ void k4_bn_relu(float* __restrict__ out,
                           const float* __restrict__ stats,
                           const float* __restrict__ gamma,
                           const float* __restrict__ beta,
                           int n4, float invN) {
    int stride = gridDim.x * blockDim.x;
    for (int i = blockIdx.x * blockDim.x + threadIdx.x; i < n4; i += stride) {
        float4 v = ((float4*)out)[i];
        int c = (i & 31) * 4;                        // feature index of v.x
        float4 r;
        float* vp = &v.x; float* rp = &r.x;
        #pragma unroll
        for (int j = 0; j < 4; ++j) {
            float s  = stats[c + j];
            float q  = stats[D_FEAT + c + j];
            float mu = s * invN;
            float var = q * invN - mu * mu;
            float rs = rsqrtf(var + BN_EPS);
            float y = (vp[j] - mu) * rs * gamma[c + j] + beta[c + j];
            rp[j] = fmaxf(y, 0.0f);
        }
        ((float4*)out)[i] = r;
    }
}

extern "C" void kernel_launch(void* const* d_in, const int* in_sizes, int n_in,
                              void* d_out, int out_size, void* d_ws, size_t ws_size,
                              hipStream_t stream) {
    const float* h     = (const float*)d_in[0];
    const float* norm  = (const float*)d_in[1];
    const float* eps   = (const float*)d_in[2];
    const float* gamma = (const float*)d_in[3];
    const float* beta  = (const float*)d_in[4];
    const int*   src   = (const int*)d_in[5];
    const int*   dst   = (const int*)d_in[6];

    const int N      = in_sizes[1];          // norm has N elements
    const int nedges = in_sizes[5];
    const int total  = N * D_FEAT;
    const int n4     = total >> 2;

    float* out   = (float*)d_out;            // doubles as hn scratch until k3
    float* neigh = (float*)d_ws;             // N*D floats
    float* stats = neigh + (size_t)total;    // 2*D floats: [sum | sumsq]

    // 1) prescale + zero
    {
        int blocks = (n4 + 255) / 256;
        if (blocks > 4096) blocks = 4096;
        k1_prescale_zero<<<blocks, 256, 0, stream>>>(h, norm, out, neigh, stats, n4);
    }
    // 2) edge scatter-add (wave-per-edge-pair, scalar index loads)
    {
        k2_edge_scatter<<<2048, 256, 0, stream>>>(out, src, dst, neigh, nedges);
    }
    // 3) combine + WMMA column stats
    {
        k3_combine_stats<<<512, 256, 0, stream>>>(out, neigh, norm, eps, out, stats, N);
    }
    // 4) BN + ReLU
    {
        int blocks = (n4 + 255) / 256;
        if (blocks > 4096) blocks = 4096;
        k4_bn_relu<<<blocks, 256, 0, stream>>>(out, stats, gamma, beta, n4, 1.0f / (float)N);
    }
}